// MinimalHSGBDH_1597727834501
// MI455X (gfx1250) — compile-verified
//
#include <hip/hip_runtime.h>
#include <hip/hip_bf16.h>

typedef float v2f __attribute__((ext_vector_type(2)));
typedef float v8f __attribute__((ext_vector_type(8)));

#define B_ 4
#define T_ 32
#define D_ 256
#define N_ 1024
#define NPART_ 4   // split-K factor for G matvecs

// ---------------------------------------------------------------- G init: 0.01*I per batch
__global__ void k_init_G(float* __restrict__ G) {
    size_t idx = (size_t)blockIdx.x * 256 + threadIdx.x;   // over B*N*N = 4M
    int within = (int)(idx & ((1u << 20) - 1));
    int n = within >> 10;
    int m = within & (N_ - 1);
    G[idx] = (n == m) ? 0.01f : 0.0f;
}

// ---------------------------------------------------------------- fp32 WMMA GEMM:
// A [128 x 256] row-major, Emat [256 x 1024] row-major -> out [128 x 1024]
// One wave per 16x16 tile; 4 waves per block (4 consecutive N-tiles).
__global__ void k_gemm_wmma(const float* __restrict__ A,
                            const float* __restrict__ Emat,
                            float* __restrict__ out) {
    const int wave  = threadIdx.x >> 5;
    const int lane  = threadIdx.x & 31;
    const int ntile = blockIdx.x * 4 + wave;   // 0..63
    const int mtile = blockIdx.y;              // 0..7
    const int mrow  = mtile * 16 + (lane & 15);
    const int ncol  = ntile * 16 + (lane & 15);
    const int khalf = (lane < 16) ? 0 : 2;     // ISA 16x4 f32 A/B layout

    v8f acc = {};
    for (int k0 = 0; k0 < D_; k0 += 4) {
        v2f a, b;
        a.x = A[mrow * D_ + k0 + khalf + 0];
        a.y = A[mrow * D_ + k0 + khalf + 1];
        b.x = Emat[(k0 + khalf + 0) * N_ + ncol];
        b.y = Emat[(k0 + khalf + 1) * N_ + ncol];
        acc = __builtin_amdgcn_wmma_f32_16x16x4_f32(
            /*neg_a=*/false, a, /*neg_b=*/false, b,
            /*c_mod=*/(short)0, acc, /*reuse_a=*/false, /*reuse_b=*/false);
    }
    // C/D layout: VGPR r, lane L -> M = r + (L<16?0:8), N = L%16
    const int mbase = mtile * 16 + ((lane < 16) ? 0 : 8);
#pragma unroll
    for (int r = 0; r < 8; ++r)
        out[(size_t)(mbase + r) * N_ + ncol] = acc[r];
}

// ---------------------------------------------------------------- layernorm + relu per row of 1024
// rows 0..127 -> x path, 128..255 -> target path
__global__ void k_ln_relu(const float* __restrict__ preX, const float* __restrict__ preT,
                          float* __restrict__ xN, float* __restrict__ tN) {
    int row = blockIdx.x;
    const float* src;
    float* dst;
    if (row < B_ * T_) { src = preX + (size_t)row * N_;            dst = xN + (size_t)row * N_; }
    else               { int r = row - B_ * T_;
                         src = preT + (size_t)r * N_;              dst = tN + (size_t)r * N_; }
    __shared__ float red[256];
    int tid = threadIdx.x;
    float v0 = src[tid], v1 = src[tid + 256], v2 = src[tid + 512], v3 = src[tid + 768];
    red[tid] = v0 + v1 + v2 + v3;
    __syncthreads();
    for (int off = 128; off; off >>= 1) { if (tid < off) red[tid] += red[tid + off]; __syncthreads(); }
    float mu = red[0] * (1.0f / N_);
    __syncthreads();
    float d0 = v0 - mu, d1 = v1 - mu, d2 = v2 - mu, d3 = v3 - mu;
    red[tid] = d0 * d0 + d1 * d1 + d2 * d2 + d3 * d3;
    __syncthreads();
    for (int off = 128; off; off >>= 1) { if (tid < off) red[tid] += red[tid + off]; __syncthreads(); }
    float inv = rsqrtf(red[0] * (1.0f / N_) + 1e-5f);
    dst[tid]       = fmaxf(d0 * inv, 0.0f);
    dst[tid + 256] = fmaxf(d1 * inv, 0.0f);
    dst[tid + 512] = fmaxf(d2 * inv, 0.0f);
    dst[tid + 768] = fmaxf(d3 * inv, 0.0f);
}

// ---------------------------------------------------------------- split-K matvec:
// partial[chunk][b][m] = sum_{n in chunk} v[b,n] * G[b,n,m]
// grid ((4 mblk)*(4 chunk), B), 256 threads. Input vector is either a plain
// row (nparts==1) or NPART_ partials (finalized here during LDS staging —
// fixed summation order, bit-deterministic).
__global__ void k_matvec_G(const float* __restrict__ vin, int vstride, int nparts,
                           const float* __restrict__ G, float* __restrict__ outp) {
    int b     = blockIdx.y;
    int mblk  = blockIdx.x & 3;
    int chunk = blockIdx.x >> 2;
    int tid   = threadIdx.x;
    int m     = mblk * 256 + tid;
    int nbase = chunk * 256;
    __shared__ float sv[256];
    float vv;
    if (nparts == 1) {
        vv = vin[(size_t)b * vstride + nbase + tid];
    } else {
        const float* v = vin + (size_t)b * N_ + nbase + tid;
        const size_t PS = (size_t)B_ * N_;
        vv = v[0] + v[PS] + v[2 * PS] + v[3 * PS];
    }
    sv[tid] = vv;
    __syncthreads();
    const float* g = G + ((size_t)b << 20) + ((size_t)nbase << 10) + m;
    float acc = 0.0f;
    for (int n0 = 0; n0 < 256; n0 += 64) {
        __builtin_prefetch(g + ((size_t)(n0 + 64) << 10), 0, 1);  // global_prefetch
#pragma unroll 8
        for (int n = n0; n < n0 + 64; ++n)
            acc = fmaf(sv[n], g[(size_t)n << 10], acc);
    }
    outp[(size_t)chunk * (B_ * N_) + (size_t)b * N_ + m] = acc;
}

// ---------------------------------------------------------------- y[b,:] = relu(max(p1,p2,p3) @ Dy)
// p1/p2/p3 arrive as NPART_ partials; finalize + max during staging.
__global__ void k_y(const float* __restrict__ p1, const float* __restrict__ p2,
                    const float* __restrict__ p3, const float* __restrict__ Dy,
                    float* __restrict__ yout) {   // yout = out + t*D
    int b  = blockIdx.x;
    int dc = threadIdx.x;   // 0..255
    __shared__ float sr[N_];
    const size_t PS = (size_t)B_ * N_;
    for (int i = dc; i < N_; i += 256) {
        size_t o = (size_t)b * N_ + i;
        float a1 = p1[o] + p1[o + PS] + p1[o + 2 * PS] + p1[o + 3 * PS];
        float a2 = p2[o] + p2[o + PS] + p2[o + 2 * PS] + p2[o + 3 * PS];
        float a3 = p3[o] + p3[o + PS] + p3[o + 2 * PS] + p3[o + 3 * PS];
        sr[i] = fmaxf(a1, fmaxf(a2, a3));
    }
    __syncthreads();
    float acc = 0.0f;
#pragma unroll 4
    for (int m = 0; m < N_; ++m)
        acc = fmaf(sr[m], Dy[m * D_ + dc], acc);
    yout[(size_t)b * T_ * D_ + dc] = fmaxf(acc, 0.0f);
}

// ---------------------------------------------------------------- G = max(G, 0.5 * xn_t (x) tn_{t+1})
// grid (N, B): one block per (b,n) row; float4 (B128) accesses.
__global__ void k_upd(float* __restrict__ G, const float* __restrict__ xN,
                      const float* __restrict__ tN, int t) {
    int b = blockIdx.y;
    int n = blockIdx.x;
    float xv = 0.5f * xN[((size_t)(b * T_ + t) << 10) + n];
    const float4* tn = (const float4*)(tN + ((size_t)(b * T_ + t + 1) << 10));
    float4* g = (float4*)(G + ((size_t)b << 20) + ((size_t)n << 10));
    int m = threadIdx.x;
    float4 gv = g[m], tv = tn[m];
    gv.x = fmaxf(gv.x, xv * tv.x);
    gv.y = fmaxf(gv.y, xv * tv.y);
    gv.z = fmaxf(gv.z, xv * tv.z);
    gv.w = fmaxf(gv.w, xv * tv.w);
    g[m] = gv;
}

extern "C" void kernel_launch(void* const* d_in, const int* in_sizes, int n_in,
                              void* d_out, int out_size, void* d_ws, size_t ws_size,
                              hipStream_t stream) {
    const float* x_seq   = (const float*)d_in[0];   // [4,32,256]
    const float* targets = (const float*)d_in[1];   // [4,32,256]
    const float* Emat    = (const float*)d_in[2];   // [256,1024]
    const float* Dy      = (const float*)d_in[3];   // [1024,256]
    float* out = (float*)d_out;                     // [4,32,256]
    float* ws  = (float*)d_ws;

    // workspace layout (floats)
    float* G    = ws;                                   // 4*1024*1024   (16 MB, L2-resident)
    float* preX = G + (size_t)B_ * N_ * N_;             // 128*1024
    float* preT = preX + (size_t)B_ * T_ * N_;          // 128*1024
    float* xN   = preT + (size_t)B_ * T_ * N_;          // 128*1024
    float* tN   = xN   + (size_t)B_ * T_ * N_;          // 128*1024
    float* p1   = tN   + (size_t)B_ * T_ * N_;          // NPART_*4*1024 partials
    float* p2   = p1 + (size_t)NPART_ * B_ * N_;
    float* p3   = p2 + (size_t)NPART_ * B_ * N_;

    k_init_G<<<dim3((B_ * N_ * N_) / 256), 256, 0, stream>>>(G);

    // projection GEMMs via fp32 WMMA (16x16x4), then layernorm+relu
    k_gemm_wmma<<<dim3(N_ / 64, (B_ * T_) / 16), 128, 0, stream>>>(x_seq,   Emat, preX);
    k_gemm_wmma<<<dim3(N_ / 64, (B_ * T_) / 16), 128, 0, stream>>>(targets, Emat, preT);
    k_ln_relu<<<dim3(2 * B_ * T_), 256, 0, stream>>>(preX, preT, xN, tN);

    // sequential scan; batches parallel inside each kernel
    const dim3 mvGrid(4 * NPART_, B_);
    for (int t = 0; t < T_; ++t) {
        k_matvec_G<<<mvGrid, 256, 0, stream>>>(xN + (size_t)t * N_, T_ * N_, 1,      G, p1);
        k_matvec_G<<<mvGrid, 256, 0, stream>>>(p1,                  N_,      NPART_, G, p2);
        k_matvec_G<<<mvGrid, 256, 0, stream>>>(p2,                  N_,      NPART_, G, p3);
        k_y<<<dim3(B_), 256, 0, stream>>>(p1, p2, p3, Dy, out + (size_t)t * D_);
        if (t < T_ - 1)
            k_upd<<<dim3(N_, B_), 256, 0, stream>>>(G, xN, tN, t);
    }
}